// GraphSAGE_45286135169725
// MI455X (gfx1250) — compile-verified
//
#include <hip/hip_runtime.h>
#include <math.h>

// ---------------------------------------------------------------------------
// GraphSAGE forward on gfx1250 (MI455X).
// Bandwidth-dominated by two neighbor gather-max passes (1.28 GB of random
// 512B-row reads, L2-resident tables). GEMMs (~10 GFLOP fp32) run on the
// matrix pipe via V_WMMA_F32_16X16X4_F32 to keep full fp32 precision.
// ---------------------------------------------------------------------------

typedef float v2f __attribute__((ext_vector_type(2)));
typedef float v8f __attribute__((ext_vector_type(8)));

#define NN      50000
#define DD      128
#define SS      25

// ---------------------------------------------------------------------------
// Generic WMMA f32 GEMM:  out[N,128] = relu?( [A1 | A2] @ W + bias )
//   A1,A2 : [N,128] row-major (A2 may be null when halves==1)
//   W     : [halves*128, 128] row-major (in_features x out_features)
// Block: 128 threads (4 waves). Each wave owns a 16-row slab; block = 64 rows.
// A tile staged in LDS (row stride 132 floats -> 16B-aligned stores, bank-
// conflict free strided reads). B streamed from L2-resident W.
// Epilogue: uniform fast path when the whole tile is in range (781/782
// blocks) to avoid per-element exec-mask guards.
// ---------------------------------------------------------------------------
template <bool RELU>
__global__ __launch_bounds__(128)
void sage_gemm_wmma(const float* __restrict__ A1,
                    const float* __restrict__ A2,
                    const float* __restrict__ W,
                    const float* __restrict__ bias,
                    float* __restrict__ out,
                    int nrows, int halves)
{
    __shared__ float As[64][132];            // 33,792 B static LDS

    const int tid    = threadIdx.x;
    const int lane   = tid & 31;
    const int waveId = tid >> 5;
    const int row0   = blockIdx.x * 64;      // block's first row
    const int m0     = row0 + waveId * 16;   // wave's first row

    v8f acc[8];
    #pragma unroll
    for (int nt = 0; nt < 8; ++nt) acc[nt] = (v8f)0.0f;

    const int mloc   = lane & 15;            // row within 16-row tile
    const int kgroup = (lane >> 4) * 2;      // lanes 16-31 hold K+2,K+3

    for (int h = 0; h < halves; ++h) {
        const float* __restrict__ Ah = (h == 0) ? A1 : A2;
        const float* __restrict__ Wh = W + (size_t)h * 128 * 128;

        // ---- cooperative stage of the 64x128 A tile into LDS ----
        __syncthreads();
        for (int i = tid; i < 64 * 32; i += 128) {
            int r  = i >> 5;
            int c4 = i & 31;
            int gr = row0 + r;
            if (gr >= nrows) gr = nrows - 1;          // clamp (stores guarded)
            float4 v = ((const float4*)(Ah + (size_t)gr * 128))[c4];
            *(float4*)&As[r][c4 * 4] = v;
        }
        __syncthreads();

        // ---- K loop, 4 at a time, 8 column tiles of 16 ----
        #pragma unroll 1
        for (int kk = 0; kk < 128; kk += 4) {
            const int kb = kk + kgroup;
            // A operand: lane holds A[m][kb], A[m][kb+1]
            v2f a;
            const float2 av = *(const float2*)&As[waveId * 16 + mloc][kb];
            a.x = av.x; a.y = av.y;

            #pragma unroll
            for (int nt = 0; nt < 8; ++nt) {
                const int n = nt * 16 + mloc;
                // B operand: lane holds W[kb][n], W[kb+1][n]
                v2f b;
                b.x = Wh[(size_t)kb * 128 + n];
                b.y = Wh[(size_t)(kb + 1) * 128 + n];
                acc[nt] = __builtin_amdgcn_wmma_f32_16x16x4_f32(
                    false, a, false, b, (short)0, acc[nt], false, false);
            }
        }
    }

    // ---- epilogue: bias, relu, store ----
    // C/D layout: VGPR r -> row r + 8*(lane/16), column = nt*16 + lane%16.
    const int rowAdd = 8 * (lane >> 4);
    if (m0 + 16 <= nrows) {
        // Fast path: whole tile in range, no per-element guards.
        #pragma unroll
        for (int nt = 0; nt < 8; ++nt) {
            const int col = nt * 16 + mloc;
            const float bv = bias[col];
            float* __restrict__ op = out + (size_t)(m0 + rowAdd) * 128 + col;
            #pragma unroll
            for (int r = 0; r < 8; ++r) {
                float v = acc[nt][r] + bv;
                if (RELU) v = fmaxf(v, 0.0f);
                op[(size_t)r * 128] = v;
            }
        }
    } else {
        #pragma unroll
        for (int nt = 0; nt < 8; ++nt) {
            const int col = nt * 16 + mloc;
            const float bv = bias[col];
            #pragma unroll
            for (int r = 0; r < 8; ++r) {
                const int row = m0 + r + rowAdd;
                if (row < nrows) {
                    float v = acc[nt][r] + bv;
                    if (RELU) v = fmaxf(v, 0.0f);
                    out[(size_t)row * 128 + col] = v;
                }
            }
        }
    }
}

// ---------------------------------------------------------------------------
// Neighbor gather + max:  out[i,:] = max_{j<S} h[idx[i,j], :]
// One wave per node; lane handles 4 floats (32*4 = 128). Index loads are
// wave-uniform -> scalar loads; row reads are coalesced 512B bursts from L2.
// ---------------------------------------------------------------------------
__global__ __launch_bounds__(256)
void sage_gather_max(const float* __restrict__ h,
                     const int* __restrict__ nidx,
                     float* __restrict__ out,
                     int nrows)
{
    const int node = blockIdx.x * 8 + (threadIdx.x >> 5);
    const int lane = threadIdx.x & 31;
    if (node >= nrows) return;

    float4 acc = make_float4(-INFINITY, -INFINITY, -INFINITY, -INFINITY);
    const int* __restrict__ row = nidx + (size_t)node * SS;
    #pragma unroll 5
    for (int j = 0; j < SS; ++j) {
        const int idx = row[j];                       // wave-uniform -> s_load
        float4 v = ((const float4*)(h + (size_t)idx * 128))[lane];
        acc.x = fmaxf(acc.x, v.x);
        acc.y = fmaxf(acc.y, v.y);
        acc.z = fmaxf(acc.z, v.z);
        acc.w = fmaxf(acc.w, v.w);
    }
    ((float4*)(out + (size_t)node * 128))[lane] = acc;
}

// ---------------------------------------------------------------------------
// BatchNorm batch statistics: per-column sum & sum-of-squares.
// ---------------------------------------------------------------------------
__global__ void sage_zero_stats(float* __restrict__ stats)
{
    if (threadIdx.x < 256) stats[threadIdx.x] = 0.0f;  // sum[128], sumsq[128]
}

__global__ __launch_bounds__(128)
void sage_col_stats(const float* __restrict__ x,
                    float* __restrict__ sum,
                    float* __restrict__ sumsq,
                    int nrows, int rows_per_block)
{
    const int c  = threadIdx.x;                        // column 0..127
    const int r0 = blockIdx.x * rows_per_block;
    float s = 0.0f, sq = 0.0f;
    for (int r = r0; r < r0 + rows_per_block && r < nrows; ++r) {
        const float v = x[(size_t)r * 128 + c];
        s  += v;
        sq += v * v;
    }
    atomicAdd(&sum[c], s);
    atomicAdd(&sumsq[c], sq);
}

__global__ void sage_bn_finalize(const float* __restrict__ sum,
                                 const float* __restrict__ sumsq,
                                 const float* __restrict__ gamma,
                                 const float* __restrict__ beta,
                                 float* __restrict__ scale,
                                 float* __restrict__ shift,
                                 int nrows)
{
    const int c = threadIdx.x;
    const float invN = 1.0f / (float)nrows;
    const float mean = sum[c] * invN;
    const float var  = sumsq[c] * invN - mean * mean;
    const float sc   = gamma[c] * rsqrtf(var + 1e-5f);
    scale[c] = sc;
    shift[c] = beta[c] - mean * sc;
}

// ---------------------------------------------------------------------------
// Fused BatchNorm apply + row L2 normalize. One wave per row.
// ---------------------------------------------------------------------------
__global__ __launch_bounds__(256)
void sage_bn_rownorm(const float* __restrict__ xpre,
                     const float* __restrict__ scale,
                     const float* __restrict__ shift,
                     float* __restrict__ xout,
                     int nrows)
{
    const int row  = blockIdx.x * 8 + (threadIdx.x >> 5);
    const int lane = threadIdx.x & 31;
    if (row >= nrows) return;

    float4 xp = ((const float4*)(xpre + (size_t)row * 128))[lane];
    float4 sc = ((const float4*)scale)[lane];
    float4 sh = ((const float4*)shift)[lane];
    float4 v;
    v.x = xp.x * sc.x + sh.x;
    v.y = xp.y * sc.y + sh.y;
    v.z = xp.z * sc.z + sh.z;
    v.w = xp.w * sc.w + sh.w;

    float s = v.x * v.x + v.y * v.y + v.z * v.z + v.w * v.w;
    #pragma unroll
    for (int off = 16; off >= 1; off >>= 1)
        s += __shfl_xor(s, off, 32);                   // wave32 reduction

    const float inv = 1.0f / (sqrtf(s) + 1e-6f);
    v.x *= inv; v.y *= inv; v.z *= inv; v.w *= inv;
    ((float4*)(xout + (size_t)row * 128))[lane] = v;
}

// ---------------------------------------------------------------------------
extern "C" void kernel_launch(void* const* d_in, const int* in_sizes, int n_in,
                              void* d_out, int out_size, void* d_ws, size_t ws_size,
                              hipStream_t stream)
{
    const float* features = (const float*)d_in[0];
    const int*   idx1     = (const int*)  d_in[1];
    const int*   idx2     = (const int*)  d_in[2];
    const float* agg1_W   = (const float*)d_in[3];
    const float* agg1_b   = (const float*)d_in[4];
    const float* fc1_W    = (const float*)d_in[5];
    const float* fc1_b    = (const float*)d_in[6];
    const float* agg2_W   = (const float*)d_in[7];
    const float* agg2_b   = (const float*)d_in[8];
    const float* fc2_W    = (const float*)d_in[9];
    const float* fc2_b    = (const float*)d_in[10];
    const float* bn_gamma = (const float*)d_in[11];
    const float* bn_beta  = (const float*)d_in[12];

    const int n = NN;
    const size_t buf = (size_t)n * 128;
    float* bufA  = (float*)d_ws;            // h1 -> x
    float* bufB  = bufA + buf;              // agg1 -> h2
    float* bufC  = bufB + buf;              // x_pre -> agg2
    float* stats = bufC + buf;              // sum|sumsq|scale|shift (512 f)
    float* s_sum = stats, *s_sq = stats + 128, *s_sc = stats + 256, *s_sh = stats + 384;

    const dim3 gemmGrid((n + 63) / 64), gemmBlk(128);
    const dim3 waveGrid((n + 7) / 8),  waveBlk(256);
    const int  rows_per_block = 128;
    const dim3 statGrid((n + rows_per_block - 1) / rows_per_block), statBlk(128);

    // ---- layer 1 ----
    sage_gemm_wmma<true><<<gemmGrid, gemmBlk, 0, stream>>>(
        features, nullptr, agg1_W, agg1_b, bufA, n, 1);             // h1
    sage_gather_max<<<waveGrid, waveBlk, 0, stream>>>(bufA, idx1, bufB, n);
    sage_gemm_wmma<true><<<gemmGrid, gemmBlk, 0, stream>>>(
        features, bufB, fc1_W, fc1_b, bufC, n, 2);                  // x_pre
    sage_zero_stats<<<1, 256, 0, stream>>>(stats);
    sage_col_stats<<<statGrid, statBlk, 0, stream>>>(bufC, s_sum, s_sq, n, rows_per_block);
    sage_bn_finalize<<<1, 128, 0, stream>>>(s_sum, s_sq, bn_gamma, bn_beta, s_sc, s_sh, n);
    sage_bn_rownorm<<<waveGrid, waveBlk, 0, stream>>>(bufC, s_sc, s_sh, bufA, n); // x
    // ---- layer 2 ----
    sage_gemm_wmma<true><<<gemmGrid, gemmBlk, 0, stream>>>(
        bufA, nullptr, agg2_W, agg2_b, bufB, n, 1);                 // h2
    sage_gather_max<<<waveGrid, waveBlk, 0, stream>>>(bufB, idx2, bufC, n);
    sage_gemm_wmma<false><<<gemmGrid, gemmBlk, 0, stream>>>(
        bufA, bufC, fc2_W, fc2_b, (float*)d_out, n, 2);             // out
}